// MPNNMakeMessage_11330123727328
// MI455X (gfx1250) — compile-verified
//
#include <hip/hip_runtime.h>

// Sizes from the reference
#define BB 16
#define NN 128
#define HD 128
#define ED 64
#define MD 128

typedef __attribute__((ext_vector_type(2))) float v2f;
typedef __attribute__((ext_vector_type(4))) float v4f;
typedef __attribute__((ext_vector_type(8))) float v8f;
typedef __attribute__((ext_vector_type(4))) int   v4i;

#define AS1 __attribute__((address_space(1)))
#define AS3 __attribute__((address_space(3)))

#if __has_builtin(__builtin_amdgcn_global_load_async_to_lds_b128)
#define HAVE_ASYNC_LD 1
#endif
#if __has_builtin(__builtin_amdgcn_global_store_async_from_lds_b128)
#define HAVE_ASYNC_ST 1
#endif

__device__ __forceinline__ void cp16_async(const float* g, float* l) {
#ifdef HAVE_ASYNC_LD
  __builtin_amdgcn_global_load_async_to_lds_b128(
      (AS1 v4i*)(AS1 void*)g, (AS3 v4i*)(AS3 void*)l, 0, 0);
#else
  *(v4f*)l = *(const v4f*)g;
#endif
}

__device__ __forceinline__ void st16_async(const float* l, float* g) {
#ifdef HAVE_ASYNC_ST
  __builtin_amdgcn_global_store_async_from_lds_b128(
      (AS1 v4i*)(AS1 void*)g, (AS3 v4i*)(AS3 void*)l, 0, 0);
#else
  *(v4f*)g = *(const v4f*)l;
#endif
}

__device__ __forceinline__ void async_wait0() {
#if defined(HAVE_ASYNC_LD) || defined(HAVE_ASYNC_ST)
#if __has_builtin(__builtin_amdgcn_s_wait_asynccnt)
  __builtin_amdgcn_s_wait_asynccnt(0);
#else
  asm volatile("s_wait_asynccnt 0" ::: "memory");
#endif
#endif
}

// ---------------------------------------------------------------------------
// Kernel C: transpose the weight slabs once.
//   WT12[sel][m][k] = W[sel*128 + k][m]   (sel = 0,1;  2 x 128 x 128)
//   W3T [m][k]      = W[256 + k][m]       (128 x 64)
// grid = 160 x 256 threads, one element each (40960 total).
// ---------------------------------------------------------------------------
__global__ __launch_bounds__(256) void wtrans_kernel(const float* __restrict__ W,
                                                     float* __restrict__ WT12,
                                                     float* __restrict__ W3T) {
  int o = blockIdx.x * 256 + threadIdx.x;
  if (o < 2 * HD * MD) {
    int sel = o >> 14;
    int r   = o & 16383;
    int m   = r >> 7;
    int k   = r & 127;
    WT12[o] = W[((size_t)sel * HD + k) * MD + m];
  } else {
    int r = o - 2 * HD * MD;         // 0..8191
    int m = r >> 6, k = r & 63;
    W3T[r] = W[((size_t)2 * HD + k) * MD + m];
  }
}

// ---------------------------------------------------------------------------
// Kernel A: T1[b,j,m] = h[b,j,:].W1 + bias[m] ; T2[b,i,m] = h[b,i,:].W2
// grid = B*2 (x&1 selects W1-with-bias vs W2), 256 threads = 8 waves.
// ---------------------------------------------------------------------------
__global__ __launch_bounds__(256) void t12_kernel(const float* __restrict__ h,
                                                  const float* __restrict__ WT12,
                                                  const float* __restrict__ bias,
                                                  float* __restrict__ T1,
                                                  float* __restrict__ T2) {
  constexpr int LDH = HD + 4;               // bank stride 4, 16B-aligned rows
  constexpr int LDK = HD + 4;
  __shared__ float sH [NN * LDH];           // 67584 B
  __shared__ float sWT[MD * LDK];           // 67584 B

  const int bb  = blockIdx.x >> 1;
  const int sel = blockIdx.x & 1;
  const int tid = threadIdx.x, lane = tid & 31, wv = tid >> 5;

  const float* hSrc = h    + (size_t)bb  * NN * HD;
  const float* wSrc = WT12 + (size_t)sel * MD * HD;   // [m][k], 128 rows of 128

#pragma unroll
  for (int it = 0; it < 16; ++it) {
    int v = tid + it * 256;                 // vec4 index
    int row = v >> 5;                       // 32 vec4 per 128-float row
    int c4  = (v & 31) << 2;
    cp16_async(hSrc + row * HD + c4, &sH[row * LDH + c4]);
  }
#pragma unroll
  for (int it = 0; it < 16; ++it) {
    int v = tid + it * 256;
    int row = v >> 5;
    int c4  = (v & 31) << 2;
    cp16_async(wSrc + row * HD + c4, &sWT[row * LDK + c4]);
  }
  async_wait0();
  __syncthreads();

  v8f acc[8] = {};
  const int j0   = wv * 16;
  const int arow = (j0 + (lane & 15)) * LDH;
  const int koff = (lane >> 4) << 1;        // lanes 16-31 hold K+2,K+3
  const int bn   = lane & 15;

  for (int ks = 0; ks < 32; ++ks) {         // K = 128, 4 per WMMA
    int k0 = ks * 4;
    v2f a = *(const v2f*)&sH[arow + k0 + koff];
#pragma unroll
    for (int t = 0; t < 8; ++t) {
      v2f b = *(const v2f*)&sWT[(t * 16 + bn) * LDK + k0 + koff];
      acc[t] = __builtin_amdgcn_wmma_f32_16x16x4_f32(
          false, a, false, b, (short)0, acc[t], false, false);
    }
  }

  float* dst = (sel == 0 ? T1 : T2) + (size_t)bb * NN * MD;
  const int jrow = j0 + ((lane >> 4) << 3); // VGPR v -> row jrow+v
#pragma unroll
  for (int t = 0; t < 8; ++t) {
    int m = t * 16 + bn;
    float bv = (sel == 0) ? bias[m] : 0.0f;
#pragma unroll
    for (int v = 0; v < 8; ++v) {
      dst[(size_t)(jrow + v) * MD + m] = acc[t][v] + bv;
    }
  }
}

// ---------------------------------------------------------------------------
// Kernel B: one WG per (b,i).
//   out[b,i,j,m] = (e[b,i,j,:].W3 + T1[b,j,m] + T2[b,i,m]) * adj[b,i,j]
// GEMM (8 waves x 8 tiles, K=64) -> LDS-staged fused epilogue -> async store.
// ---------------------------------------------------------------------------
struct GemmBufs {
  float e [NN * (ED + 4)];                  // 34816 B
  float wt[MD * (ED + 4)];                  // 34816 B
};
union SmemB {
  GemmBufs g;
  float stage[NN * (MD + 4)];               // 67584 B  (reused after GEMM)
};

__global__ __launch_bounds__(256) void message_kernel(const float* __restrict__ e,
                                                      const float* __restrict__ adj,
                                                      const float* __restrict__ W3T,
                                                      const float* __restrict__ T1,
                                                      const float* __restrict__ T2,
                                                      float* __restrict__ out) {
  constexpr int LDE = ED + 4;
  constexpr int LDK = ED + 4;
  constexpr int LDS = MD + 4;               // stage row stride (132)
  __shared__ SmemB smem;
  __shared__ float sT2[MD];
  __shared__ float sAdj[NN];

  const int bi = blockIdx.x;                // 0..2047
  const int bb = bi >> 7, ii = bi & 127;
  const int tid = threadIdx.x, lane = tid & 31, wv = tid >> 5;

  float* sE  = smem.g.e;
  float* sWT = smem.g.wt;

  const float* eSrc = e + ((size_t)bb * NN + ii) * (NN * ED);

  // Stage e[b,i] (128x64) and W3T (128x64) via async 128b copies.
#pragma unroll
  for (int it = 0; it < 8; ++it) {
    int v = tid + it * 256;
    int row = v >> 4;                       // 16 vec4 per 64-float row
    int c4  = (v & 15) << 2;
    cp16_async(eSrc + row * ED + c4, &sE[row * LDE + c4]);
  }
#pragma unroll
  for (int it = 0; it < 8; ++it) {
    int v = tid + it * 256;
    int row = v >> 4;
    int c4  = (v & 15) << 2;
    cp16_async(W3T + row * ED + c4, &sWT[row * LDK + c4]);
  }
  if (tid < MD) {
    sT2[tid] = T2[((size_t)bb * NN + ii) * MD + tid];
  } else if (tid < MD + NN) {
    int j = tid - MD;
    sAdj[j] = adj[((size_t)bb * NN + ii) * NN + j];
  }
  async_wait0();
  __syncthreads();

  v8f acc[8] = {};
  const int j0   = wv * 16;
  const int arow = (j0 + (lane & 15)) * LDE;
  const int koff = (lane >> 4) << 1;
  const int bn   = lane & 15;

  for (int ks = 0; ks < 16; ++ks) {         // K = 64
    int k0 = ks * 4;
    v2f a = *(const v2f*)&sE[arow + k0 + koff];
#pragma unroll
    for (int t = 0; t < 8; ++t) {
      v2f b = *(const v2f*)&sWT[(t * 16 + bn) * LDK + k0 + koff];
      acc[t] = __builtin_amdgcn_wmma_f32_16x16x4_f32(
          false, a, false, b, (short)0, acc[t], false, false);
    }
  }
  __syncthreads();                          // sE/sWT dead; stage takes over

  // Phase 1: bulk async-load T1[b] (128x128) into stage (L2-resident source).
  const float* t1b = T1 + (size_t)bb * NN * MD;
  float* stage = smem.stage;
#pragma unroll
  for (int it = 0; it < 16; ++it) {
    int v = tid + it * 256;
    int row = v >> 5;                       // 32 vec4 per 128-float row
    int c4  = (v & 31) << 2;
    cp16_async(t1b + row * MD + c4, &stage[row * LDS + c4]);
  }
  async_wait0();
  __syncthreads();

  // Phase 2: fuse (acc + t1 + t2) * adj in place in LDS.
  const int jrow = j0 + ((lane >> 4) << 3);
#pragma unroll
  for (int t = 0; t < 8; ++t) {
    int m = t * 16 + bn;
    float t2v = sT2[m];
#pragma unroll
    for (int v = 0; v < 8; ++v) {
      int j = jrow + v;
      float* p = &stage[j * LDS + m];
      *p = (acc[t][v] + *p + t2v) * sAdj[j];
    }
  }
  __syncthreads();

  // Phase 3: stream LDS -> out with async b128 stores (512B/wave, no VGPRs).
  float* outB = out + ((size_t)bb * NN + ii) * (size_t)(NN * MD);
#pragma unroll
  for (int it = 0; it < 16; ++it) {
    int v = tid + it * 256;
    int row = v >> 5;
    int c4  = (v & 31) << 2;
    st16_async(&stage[row * LDS + c4], outB + (size_t)row * MD + c4);
  }
  // S_ENDPGM performs an implicit wait-idle, covering outstanding async stores.
}

// ---------------------------------------------------------------------------
// Host launch.  Workspace (needs ~2.2 MB):
//   T1 (1 MB) | T2 (1 MB) | WT12 (128 KB) | W3T (32 KB)
// All fully rewritten every call (deterministic).
// ---------------------------------------------------------------------------
extern "C" void kernel_launch(void* const* d_in, const int* in_sizes, int n_in,
                              void* d_out, int out_size, void* d_ws, size_t ws_size,
                              hipStream_t stream) {
  const float* h    = (const float*)d_in[0];
  const float* e    = (const float*)d_in[1];
  const float* adj  = (const float*)d_in[2];
  const float* W    = (const float*)d_in[3];
  const float* bias = (const float*)d_in[4];
  float* out = (float*)d_out;

  float* T1   = (float*)d_ws;
  float* T2   = T1 + (size_t)BB * NN * MD;
  float* WT12 = T2 + (size_t)BB * NN * MD;
  float* W3T  = WT12 + (size_t)2 * HD * MD;

  wtrans_kernel <<<dim3(160),     dim3(256), 0, stream>>>(W, WT12, W3T);
  t12_kernel    <<<dim3(BB * 2),  dim3(256), 0, stream>>>(h, WT12, bias, T1, T2);
  message_kernel<<<dim3(BB * NN), dim3(256), 0, stream>>>(e, adj, W3T, T1, T2, out);
}